// BoundNoiseSampler_79937931313210
// MI455X (gfx1250) — compile-verified
//
#include <hip/hip_runtime.h>
#include <hip/hip_bf16.h>

typedef __attribute__((ext_vector_type(4))) float v4f;

__device__ __forceinline__ float bound_noise_weight(float s) {
    // Constants folded: sigma_data=0.5, R=7 -> 4R^2=196, T=80, delta=0.5, norm=2
    //   karras = (sig2+0.25)/(0.25*sig2) = 4 + 1/sig2
    //   out    = 4 + (1/sig2) * (1 + exp(-0.5*inv_C*sig2))
    //   C      = 6*(196+sig2)*exp(196/sig2);  inv_C = rcp(C) (rcp(inf)=0 == reference)
    const float K_POS = 282.76822801423683f;   //  196 * log2(e)
    const float K_NEG = -0.7213475204444817f;  // -0.5 * log2(e)

    float sig2     = s * s;
    float inv_sig2 = __builtin_amdgcn_rcpf(sig2);              // v_rcp_f32

    // e_pos = exp(196/sig2) = exp2(K_POS * inv_sig2)
    float e_pos = __builtin_amdgcn_exp2f(K_POS * inv_sig2);    // v_exp_f32

    // C = 6*(196+sig2)*e_pos ; rcp maps overflow->0 exactly like the reference mask
    float C     = 6.0f * (196.0f + sig2) * e_pos;
    float inv_C = __builtin_amdgcn_rcpf(C);                    // v_rcp_f32, rcp(inf)=+0

    // e_neg = exp(-0.5 * inv_C * sig2) = exp2(K_NEG * inv_C * sig2)
    float e_neg = __builtin_amdgcn_exp2f((K_NEG * inv_C) * sig2);  // v_exp_f32

    // out = 4 + inv_sig2 + inv_sig2 * e_neg
    return __builtin_fmaf(inv_sig2, e_neg, inv_sig2 + 4.0f);
}

__launch_bounds__(256)
__global__ void BoundNoiseSampler_79937931313210_kernel(const float* __restrict__ sigma,
                                                        float* __restrict__ out,
                                                        int n) {
    int tid  = blockIdx.x * blockDim.x + threadIdx.x;
    int base = tid * 4;

    if (base + 3 < n) {
        // 16B/lane non-temporal load -> global_load_b128 th:NT (512 B per wave32)
        v4f s = __builtin_nontemporal_load((const v4f*)(sigma + base));

        v4f r;
        r.x = bound_noise_weight(s.x);
        r.y = bound_noise_weight(s.y);
        r.z = bound_noise_weight(s.z);
        r.w = bound_noise_weight(s.w);

        // Non-temporal store -> global_store_b128 th:NT: 256 MB single-touch
        // stream must not evict the 192 MB L2's useful contents
        __builtin_nontemporal_store(r, (v4f*)(out + base));
    } else if (base < n) {
        // Scalar tail (N = 2^25 is a multiple of 4; kept for robustness)
        for (int i = base; i < n; ++i) {
            out[i] = bound_noise_weight(sigma[i]);
        }
    }
}

extern "C" void kernel_launch(void* const* d_in, const int* in_sizes, int n_in,
                              void* d_out, int out_size, void* d_ws, size_t ws_size,
                              hipStream_t stream) {
    const float* sigma = (const float*)d_in[0];
    float* out = (float*)d_out;
    int n = in_sizes[0];

    // One float4 per thread: ceil(n/4) work items, 256 threads (8 wave32) per block.
    int n_vec4 = (n + 3) / 4;
    int threads = 256;
    int blocks = (n_vec4 + threads - 1) / threads;

    BoundNoiseSampler_79937931313210_kernel<<<blocks, threads, 0, stream>>>(sigma, out, n);
}